// MobileFacenetUnsupervisedLoss_58634893525570
// MI455X (gfx1250) — compile-verified
//
#include <hip/hip_runtime.h>

// Problem constants (match reference)
#define NROWS 8192
#define KCOLS 8192
#define DIM   128

// Fused-kernel tiling
#define MT       2              // 16-row tiles per block -> 32 rows/block
#define BLKROWS  (MT * 16)
#define WAVES    8              // 256 threads = 8 wave32

typedef __attribute__((ext_vector_type(2))) float v2f;
typedef __attribute__((ext_vector_type(8))) float v8f;

#define NEG_INF (-3.40282347e38f)

// ---------------------------------------------------------------------------
// ws init: zero the atomic accumulators (colsum[DIM], intra[1])
// ---------------------------------------------------------------------------
__global__ void mful_init(float* __restrict__ colsum, float* __restrict__ intra) {
  int t = threadIdx.x;
  if (t < DIM) colsum[t] = 0.0f;
  if (t == DIM) intra[0] = 0.0f;
}

// ---------------------------------------------------------------------------
// One wave per row: squared L2 norm (float4 loads + shfl_xor reduce, wave32)
// ---------------------------------------------------------------------------
__global__ void mful_row_sq(const float* __restrict__ a, float* __restrict__ out, int rows) {
  int row  = (int)((blockIdx.x * blockDim.x + threadIdx.x) >> 5);
  int lane = threadIdx.x & 31;
  if (row >= rows) return;
  const float4* p = (const float4*)(a + (size_t)row * DIM);
  float4 v = p[lane];                         // 32 lanes * 4 = 128 = DIM
  float s = v.x * v.x + v.y * v.y + v.z * v.z + v.w * v.w;
#pragma unroll
  for (int i = 16; i >= 1; i >>= 1) s += __shfl_xor(s, i, 32);
  if (lane == 0) out[row] = s;
}

// ---------------------------------------------------------------------------
// invs[k] = 1 / s[k]
// ---------------------------------------------------------------------------
__global__ void mful_recip(const float* __restrict__ s, float* __restrict__ invs) {
  int i = (int)(blockIdx.x * blockDim.x + threadIdx.x);
  if (i < KCOLS) invs[i] = 1.0f / s[i];
}

// ---------------------------------------------------------------------------
// Column sums of m (for the inter term). 128 threads/block (thread d == dim d,
// coalesced), each block covers 256 rows, one atomic per dim per block.
// ---------------------------------------------------------------------------
__global__ void mful_colsum(const float* __restrict__ m, float* __restrict__ colsum) {
  int d  = threadIdx.x;                 // 0..127
  int k0 = (int)blockIdx.x * 256;
  float acc = 0.0f;
  for (int k = 0; k < 256; ++k) acc += m[(size_t)(k0 + k) * DIM + d];
  atomicAdd(&colsum[d], acc);
}

// ---------------------------------------------------------------------------
// Fused: WMMA f32 GEMM tile (x @ m^T) -> distance -> logits -> online
// log-sum-exp + entropy accumulation. One block = 32 rows of x; 8 waves
// partition the K=512 column-chunks; A-fragments stay in VGPRs all loop long.
// ---------------------------------------------------------------------------
__global__ void __launch_bounds__(256, 1)
mful_fused(const float* __restrict__ x, const float* __restrict__ m,
           const float* __restrict__ x2, const float* __restrict__ m2,
           const float* __restrict__ invs, float* __restrict__ intra) {
  __shared__ float sM[WAVES][MT][8][32];
  __shared__ float sZ[WAVES][MT][8][32];
  __shared__ float sS[WAVES][MT][8][32];

  const int lane = threadIdx.x & 31;
  const int wave = threadIdx.x >> 5;
  const int hf   = lane >> 4;           // which half-wave (K split for f32 frags)
  const int ln   = lane & 15;
  const int rowbase = (int)blockIdx.x * BLKROWS;

  // A fragments: 16x4 f32 per WMMA step. Lane ln holds row (rowbase+mt*16+ln);
  // lanes 0-15 carry K=d..d+1, lanes 16-31 carry K=d+2..d+3 (ISA 32-bit A layout).
  v2f af[MT][DIM / 4];
#pragma unroll
  for (int t = 0; t < DIM / 4; ++t) {
    const int d = t * 4 + hf * 2;
#pragma unroll
    for (int mt = 0; mt < MT; ++mt) {
      const float* p = x + (size_t)(rowbase + mt * 16 + ln) * DIM + d;
      af[mt][t] = *(const v2f*)p;
    }
  }

  // x squared norms for the rows this lane's C slots correspond to
  // (C/D layout: VGPR j -> row j for lanes 0-15, row j+8 for lanes 16-31)
  float x2r[MT][8];
#pragma unroll
  for (int mt = 0; mt < MT; ++mt)
#pragma unroll
    for (int j = 0; j < 8; ++j)
      x2r[mt][j] = x2[rowbase + mt * 16 + hf * 8 + j];

  // Online softmax-entropy state per C slot
  float Mx[MT][8], Zz[MT][8], Ss[MT][8];
#pragma unroll
  for (int mt = 0; mt < MT; ++mt)
#pragma unroll
    for (int j = 0; j < 8; ++j) { Mx[mt][j] = NEG_INF; Zz[mt][j] = 0.0f; Ss[mt][j] = 0.0f; }

  const v8f vzero = {0.f, 0.f, 0.f, 0.f, 0.f, 0.f, 0.f, 0.f};

  // Wave-uniform K loop (EXEC stays all-ones for WMMA)
  for (int kc = wave; kc < KCOLS / 16; kc += WAVES) {
    const int col  = kc * 16 + ln;
    const float m2c = m2[col];
    const float isc = invs[col];

    v8f c[MT];
#pragma unroll
    for (int mt = 0; mt < MT; ++mt) c[mt] = vzero;

#pragma unroll
    for (int t = 0; t < DIM / 4; ++t) {
      const int d = t * 4 + hf * 2;
      // B fragment (4x16): lane ln holds column (kc*16+ln); K split across halves
      v2f bf = *(const v2f*)(m + (size_t)col * DIM + d);
#pragma unroll
      for (int mt = 0; mt < MT; ++mt)
        c[mt] = __builtin_amdgcn_wmma_f32_16x16x4_f32(
            false, af[mt][t], false, bf, (short)0, c[mt], false, false);
    }

    // Epilogue: distance -> logit -> online (M, Z, S) update
#pragma unroll
    for (int mt = 0; mt < MT; ++mt) {
#pragma unroll
      for (int j = 0; j < 8; ++j) {
        float acc  = c[mt][j];
        float d2   = x2r[mt][j] + m2c - 2.0f * acc;
        float dist = __builtin_sqrtf(fmaxf(d2, 0.0f));
        float l    = -dist * isc;
        float Mo   = Mx[mt][j];
        float Mn   = fmaxf(Mo, l);
        float sc   = __expf(Mo - Mn);   // 0 on first chunk (exp(-inf))
        float e    = __expf(l - Mn);
        Zz[mt][j]  = Zz[mt][j] * sc + e;
        Ss[mt][j]  = Ss[mt][j] * sc + e * l;
        Mx[mt][j]  = Mn;
      }
    }
  }

  // Spill per-lane state to LDS and merge
#pragma unroll
  for (int mt = 0; mt < MT; ++mt)
#pragma unroll
    for (int j = 0; j < 8; ++j) {
      sM[wave][mt][j][lane] = Mx[mt][j];
      sZ[wave][mt][j][lane] = Zz[mt][j];
      sS[wave][mt][j][lane] = Ss[mt][j];
    }
  __syncthreads();

  if (threadIdx.x < BLKROWS) {        // 32 threads, one per row (full wave 0)
    int r  = (int)threadIdx.x;
    int mt = r >> 4;
    int rh = (r >> 3) & 1;            // which lane-half held this row
    int j  = r & 7;
    float M = NEG_INF, Z = 0.0f, S = 0.0f;
    for (int w = 0; w < WAVES; ++w) {
      for (int l2 = 0; l2 < 16; ++l2) {
        int li = rh * 16 + l2;
        float Mp = sM[w][mt][j][li];
        float Zp = sZ[w][mt][j][li];
        float Sp = sS[w][mt][j][li];
        float Mn = fmaxf(M, Mp);
        float s1 = __expf(M - Mn);
        float s2 = __expf(Mp - Mn);
        Z = Z * s1 + Zp * s2;
        S = S * s1 + Sp * s2;
        M = Mn;
      }
    }
    float ent = M + __logf(Z) - S / Z;   // per-row entropy
#pragma unroll
    for (int i = 16; i >= 1; i >>= 1) ent += __shfl_xor(ent, i, 32);
    if (threadIdx.x == 0) atomicAdd(intra, ent);
  }
}

// ---------------------------------------------------------------------------
// Finalize: out = intra/N - (sum(m2)/K - ||colsum/K||^2)
// ---------------------------------------------------------------------------
__global__ void mful_finalize(const float* __restrict__ m2,
                              const float* __restrict__ colsum,
                              const float* __restrict__ intra,
                              float* __restrict__ out) {
  __shared__ float red[256];
  int t = threadIdx.x;

  float a = 0.0f;
  for (int k = t; k < KCOLS; k += 256) a += m2[k];
  red[t] = a;
  __syncthreads();
  for (int s = 128; s > 0; s >>= 1) {
    if (t < s) red[t] += red[t + s];
    __syncthreads();
  }
  float sum_m2 = red[0];
  __syncthreads();

  float b = 0.0f;
  if (t < DIM) {
    float mu = colsum[t] * (1.0f / (float)KCOLS);
    b = mu * mu;
  }
  red[t] = b;
  __syncthreads();
  for (int s = 128; s > 0; s >>= 1) {
    if (t < s) red[t] += red[t + s];
    __syncthreads();
  }
  if (t == 0) {
    float inter = sum_m2 * (1.0f / (float)KCOLS) - red[0];
    out[0] = intra[0] * (1.0f / (float)NROWS) - inter;
  }
}

// ---------------------------------------------------------------------------
// Entry point
// ---------------------------------------------------------------------------
extern "C" void kernel_launch(void* const* d_in, const int* in_sizes, int n_in,
                              void* d_out, int out_size, void* d_ws, size_t ws_size,
                              hipStream_t stream) {
  const float* x = (const float*)d_in[0];   // [N, D]
  const float* m = (const float*)d_in[1];   // [K, D]
  const float* s = (const float*)d_in[2];   // [K]
  float* out = (float*)d_out;

  float* w      = (float*)d_ws;
  float* x2     = w;                  // N
  float* m2     = x2 + NROWS;         // K
  float* invs   = m2 + KCOLS;         // K
  float* colsum = invs + KCOLS;       // DIM
  float* intra  = colsum + DIM;       // 1

  mful_init<<<1, 256, 0, stream>>>(colsum, intra);
  mful_row_sq<<<NROWS / 8, 256, 0, stream>>>(x, x2, NROWS);
  mful_row_sq<<<KCOLS / 8, 256, 0, stream>>>(m, m2, KCOLS);
  mful_recip<<<KCOLS / 256, 256, 0, stream>>>(s, invs);
  mful_colsum<<<KCOLS / 256, 128, 0, stream>>>(m, colsum);
  mful_fused<<<NROWS / BLKROWS, 256, 0, stream>>>(x, m, x2, m2, invs, intra);
  mful_finalize<<<1, 256, 0, stream>>>(m2, colsum, intra, out);
}